// TwoLayerSimpleGCN_2233382994097
// MI455X (gfx1250) — compile-verified
//
#include <hip/hip_runtime.h>
#include <math.h>

// GCN forward on MI455X (gfx1250, wave32).
// Pipeline per layer: hlin = h @ W (fp32 WMMA 16x16x4), zero(agg),
// edge scatter (wave-per-edge, L2-resident atomics), finalize (self-loop +
// bias + relu). Then max/mean pooling via atomics and a tiny sigmoid head.

typedef __attribute__((ext_vector_type(2))) float v2f;
typedef __attribute__((ext_vector_type(8))) float v8f;

// ---------------------------------------------------------------- zero fill
__global__ void zero_f4(float4* __restrict__ p, int n4) {
  int i = blockIdx.x * blockDim.x + threadIdx.x;
  if (i < n4) p[i] = make_float4(0.f, 0.f, 0.f, 0.f);
}

// ------------------------------------------------------- degree / norm prep
__global__ void deg_kernel(const int* __restrict__ dst, const float* __restrict__ ew,
                           float* __restrict__ deg, int E) {
  int e = blockIdx.x * blockDim.x + threadIdx.x;
  if (e < E) atomicAdd(&deg[dst[e]], ew[e]);
}

// deg -> dinv (in place) and dinv2 = 1/(deg+1)   (deg+1 >= 1 so no zero guard)
__global__ void dinv_kernel(float* __restrict__ deg, float* __restrict__ dinv2, int N) {
  int i = blockIdx.x * blockDim.x + threadIdx.x;
  if (i < N) {
    float d = deg[i] + 1.0f;
    deg[i]  = rsqrtf(d);
    dinv2[i] = 1.0f / d;
  }
}

__global__ void norm_kernel(const int* __restrict__ src, const int* __restrict__ dst,
                            const float* __restrict__ ew, const float* __restrict__ dinv,
                            float* __restrict__ nrm, int E) {
  int e = blockIdx.x * blockDim.x + threadIdx.x;
  if (e < E) nrm[e] = dinv[src[e]] * ew[e] * dinv[dst[e]];
}

// ------------------------------------------------------------- fp32 WMMA GEMM
// Y[M x 64] = X[M x K] @ W[K x 64].  One 256-thread block = 8 waves covering a
// 32x64 output tile (2 m-subtiles x 4 n-subtiles of 16x16).  M must be a
// multiple of 32 (100000 = 3125*32), K a multiple of 4.  EXEC is all-ones.
__global__ __launch_bounds__(256)
void gemm_n64_wmma(const float* __restrict__ X, const float* __restrict__ W,
                   float* __restrict__ Y, int K) {
  const int lane = threadIdx.x & 31;
  const int wave = threadIdx.x >> 5;     // 0..7
  const int mt   = wave >> 2;            // 0..1
  const int nt   = wave & 3;             // 0..3
  const int m0   = blockIdx.x * 32 + mt * 16;
  const int n0   = nt * 16;
  const int half = lane >> 4;            // 0: K=k..k+1, 1: K=k+2..k+3
  const int l16  = lane & 15;

  const float* __restrict__ xrow = X + (size_t)(m0 + l16) * K;

  v8f c = {};
  for (int k = 0; k < K; k += 4) {
    // A fragment: lane holds A[m0+l16][k + 2*half + {0,1}] (contiguous pair)
    v2f a;
    const float* ap = xrow + k + 2 * half;
    a.x = ap[0];
    a.y = ap[1];
    // B fragment: lane holds B[k + 2*half + {0,1}][n0 + l16]
    v2f b;
    const float* bp = W + (size_t)(k + 2 * half) * 64 + n0 + l16;
    b.x = bp[0];
    b.y = bp[64];
    c = __builtin_amdgcn_wmma_f32_16x16x4_f32(
        /*neg_a=*/false, a, /*neg_b=*/false, b,
        /*c_mod=*/(short)0, c, /*reuse_a=*/false, /*reuse_b=*/false);
  }

  // D layout: VGPR r -> row m0 + r + 8*half, col n0 + l16
  float* yr = Y + (size_t)(m0 + 8 * half) * 64 + n0 + l16;
#pragma unroll
  for (int r = 0; r < 8; ++r) yr[(size_t)r * 64] = c[r];
}

// --------------------------------------------------------------- edge scatter
// One wave32 per edge, each lane owns 2 of 64 features.  h table (25.6 MB) is
// L2-resident (192 MB L2), so gather + atomics stay on-chip.
__global__ __launch_bounds__(256)
void scatter_kernel(const float* __restrict__ hlin, const int* __restrict__ src,
                    const int* __restrict__ dst, const float* __restrict__ nrm,
                    float* __restrict__ agg, int E) {
  int t = blockIdx.x * blockDim.x + threadIdx.x;
  int e = t >> 5;
  int lane = t & 31;
  if (e >= E) return;
  int s = src[e];
  int d = dst[e];
  float w = nrm[e];
  const float2 h = *(const float2*)(hlin + (size_t)s * 64 + lane * 2);
  float* out = agg + (size_t)d * 64 + lane * 2;
  atomicAdd(out + 0, w * h.x);
  atomicAdd(out + 1, w * h.y);
}

// ----------------------------------------------------- self-loop + bias + relu
__global__ void finalize_kernel(const float4* __restrict__ agg, const float4* __restrict__ hlin,
                                const float* __restrict__ dinv2, const float* __restrict__ bias,
                                float4* __restrict__ hout, int n4) {
  int i = blockIdx.x * blockDim.x + threadIdx.x;
  if (i >= n4) return;
  int node = i >> 4;           // 16 float4 per node (64 feats)
  int f    = (i & 15) * 4;
  float d2 = dinv2[node];
  float4 a = agg[i];
  float4 h = hlin[i];
  float4 r;
  r.x = fmaxf(a.x + d2 * h.x + bias[f + 0], 0.f);
  r.y = fmaxf(a.y + d2 * h.y + bias[f + 1], 0.f);
  r.z = fmaxf(a.z + d2 * h.z + bias[f + 2], 0.f);
  r.w = fmaxf(a.w + d2 * h.w + bias[f + 3], 0.f);
  hout[i] = r;
}

// ------------------------------------------------------------------- pooling
// h >= 0 after relu, so uint-bitwise atomicMax == float max; zero-init matches
// the reference's where(counts>0, gmax, 0).
__global__ __launch_bounds__(256)
void pool_kernel(const float* __restrict__ h, const int* __restrict__ batch,
                 unsigned* __restrict__ gmax, float* __restrict__ gsum,
                 float* __restrict__ cnt, int N) {
  int t = blockIdx.x * blockDim.x + threadIdx.x;
  int node = t >> 5;
  int lane = t & 31;
  if (node >= N) return;
  int g = batch[node];
  float2 v = *(const float2*)(h + (size_t)node * 64 + lane * 2);
  unsigned* mp = gmax + (size_t)g * 64 + lane * 2;
  atomicMax(mp + 0, __float_as_uint(v.x));
  atomicMax(mp + 1, __float_as_uint(v.y));
  float* sp = gsum + (size_t)g * 64 + lane * 2;
  atomicAdd(sp + 0, v.x);
  atomicAdd(sp + 1, v.y);
  if (lane == 0) atomicAdd(&cnt[g], 1.0f);
}

// ------------------------------------------------------------------ MLP head
// One wave per graph: emit hidden = [gmax | gmean], out = sigmoid(hidden@Wout+b).
__global__ __launch_bounds__(256)
void head_kernel(const float* __restrict__ gmax, const float* __restrict__ gsum,
                 const float* __restrict__ cnt, const float* __restrict__ Wout,
                 const float* __restrict__ bout, float* __restrict__ out,
                 float* __restrict__ hidden, int G) {
  int t = blockIdx.x * blockDim.x + threadIdx.x;
  int g = t >> 5;
  int lane = t & 31;
  if (g >= G) return;
  float inv = 1.0f / fmaxf(cnt[g], 1.0f);
  float2 mx = *(const float2*)(gmax + (size_t)g * 64 + lane * 2);
  float2 sm = *(const float2*)(gsum + (size_t)g * 64 + lane * 2);
  float2 mn = make_float2(sm.x * inv, sm.y * inv);
  float* hd = hidden + (size_t)g * 128;
  hd[lane * 2 + 0]      = mx.x;
  hd[lane * 2 + 1]      = mx.y;
  hd[64 + lane * 2 + 0] = mn.x;
  hd[64 + lane * 2 + 1] = mn.y;
  float p = mx.x * Wout[lane * 2] + mx.y * Wout[lane * 2 + 1] +
            mn.x * Wout[64 + lane * 2] + mn.y * Wout[64 + lane * 2 + 1];
#pragma unroll
  for (int off = 16; off > 0; off >>= 1) p += __shfl_xor(p, off, 32);
  if (lane == 0) out[g] = 1.0f / (1.0f + expf(-(p + bout[0])));
}

// =============================================================== launch host
extern "C" void kernel_launch(void* const* d_in, const int* in_sizes, int n_in,
                              void* d_out, int out_size, void* d_ws, size_t ws_size,
                              hipStream_t stream) {
  const float* x     = (const float*)d_in[0];
  const int*   ei    = (const int*)d_in[1];
  const float* ew    = (const float*)d_in[2];
  const int*   batch = (const int*)d_in[3];
  const float* W1    = (const float*)d_in[5];
  const float* b1    = (const float*)d_in[6];
  const float* W2    = (const float*)d_in[7];
  const float* b2    = (const float*)d_in[8];
  const float* W3    = (const float*)d_in[9];
  const float* b3    = (const float*)d_in[10];
  const float* Wout  = (const float*)d_in[11];
  const float* bout  = (const float*)d_in[12];

  const int nNodes = in_sizes[3];           // 100000 (multiple of 32)
  const int nEdges = in_sizes[2];           // 1600000
  const int Din    = in_sizes[0] / nNodes;  // 128
  const int G      = 512;                   // N_GRAPHS (reference constant)

  const int* srcI = ei;
  const int* dstI = ei + nEdges;

  // workspace layout (floats)
  float* ws    = (float*)d_ws;
  float* deg   = ws; ws += nNodes;                 // becomes dinv in place
  float* dinv2 = ws; ws += nNodes;
  float* nrm   = ws; ws += nEdges;
  float* hlin  = ws; ws += (size_t)nNodes * 64;
  float* agg   = ws; ws += (size_t)nNodes * 64;
  float* hbuf  = ws; ws += (size_t)nNodes * 64;
  float* gmax  = ws; ws += (size_t)G * 64;
  float* gsum  = ws; ws += (size_t)G * 64;
  float* cnt   = ws; ws += G;

  const int TB = 256;
  auto blocks = [](long n, int tb) { return (int)((n + tb - 1) / tb); };

  // zero: deg + pooling accumulators (gmax/gsum/cnt are contiguous)
  zero_f4<<<blocks(nNodes / 4, TB), TB, 0, stream>>>((float4*)deg, nNodes / 4);
  {
    int n4 = (G * 64 * 2 + G) / 4;
    zero_f4<<<blocks(n4, TB), TB, 0, stream>>>((float4*)gmax, n4);
  }

  // degree + symmetric normalization (computed once, reused by all 3 layers)
  deg_kernel<<<blocks(nEdges, TB), TB, 0, stream>>>(dstI, ew, deg, nEdges);
  dinv_kernel<<<blocks(nNodes, TB), TB, 0, stream>>>(deg, dinv2, nNodes);
  norm_kernel<<<blocks(nEdges, TB), TB, 0, stream>>>(srcI, dstI, ew, deg, nrm, nEdges);

  const int gemmBlocks = nNodes / 32;                  // 3125
  const int aggN4      = nNodes * 16;                  // nNodes*64/4
  const long scatterT  = (long)nEdges * 32;

  const float* hin = x;
  int K = Din;
  const float* Ws[3] = {W1, W2, W3};
  const float* bs[3] = {b1, b2, b3};
  for (int layer = 0; layer < 3; ++layer) {
    gemm_n64_wmma<<<gemmBlocks, TB, 0, stream>>>(hin, Ws[layer], hlin, K);
    zero_f4<<<blocks(aggN4, TB), TB, 0, stream>>>((float4*)agg, aggN4);
    scatter_kernel<<<blocks(scatterT, TB), TB, 0, stream>>>(hlin, srcI, dstI, nrm, agg, nEdges);
    finalize_kernel<<<blocks(aggN4, TB), TB, 0, stream>>>(
        (const float4*)agg, (const float4*)hlin, dinv2, bs[layer], (float4*)hbuf, aggN4);
    hin = hbuf;
    K = 64;
  }

  pool_kernel<<<blocks((long)nNodes * 32, TB), TB, 0, stream>>>(
      hbuf, batch, (unsigned*)gmax, gsum, cnt, nNodes);

  float* out    = (float*)d_out;        // [512]
  float* hidden = (float*)d_out + G;    // [512 x 128]
  head_kernel<<<blocks((long)G * 32, TB), TB, 0, stream>>>(
      gmax, gsum, cnt, Wout, bout, out, hidden, G);
}